// HungarianMatcherPolyline_86990267613642
// MI455X (gfx1250) — compile-verified
//
#include <hip/hip_runtime.h>
#include <stdint.h>

#define B_ 32
#define Q_ 500
#define N_ 200
#define K_ 16
#define C_ 92
#define TWO_K 32
#define TP_STRIDE 36          // padded row stride (floats): 144B, 16B aligned, kills bank aliasing
#define QT 8                  // queries per block (one per wave32)
#define BLOCK 256
#define EPS_ 1e-6f

// ---- CDNA5 async global->LDS copies (ISA ch.08, ASYNCcnt-tracked) -----------
__device__ __forceinline__ void async_ld_b128(uint32_t lds_off, uint32_t goff, const void* gbase) {
    asm volatile("global_load_async_to_lds_b128 %0, %1, %2"
                 :: "v"(lds_off), "v"(goff), "s"(gbase) : "memory");
}
__device__ __forceinline__ void async_ld_b32(uint32_t lds_off, uint32_t goff, const void* gbase) {
    asm volatile("global_load_async_to_lds_b32 %0, %1, %2"
                 :: "v"(lds_off), "v"(goff), "s"(gbase) : "memory");
}
__device__ __forceinline__ void wait_async_zero() {
    asm volatile("s_wait_asynccnt 0" ::: "memory");
}

__global__ __launch_bounds__(BLOCK)
void HungarianMatcherPolyline_kernel(const float* __restrict__ pred_logits,   // (B,Q,C)
                                     const float* __restrict__ pred_poly,     // (B,Q,K,2)
                                     const int*   __restrict__ tgt_labels,    // (B,N)
                                     const float* __restrict__ tgt_poly,      // (B,N,K,2)
                                     float* __restrict__ out)                 // (B,Q,N)
{
    __shared__ float s_tp[N_ * TP_STRIDE];   // target polylines, padded rows  (28.8 KB)
    __shared__ float s_bx0[N_], s_by0[N_], s_bx1[N_], s_by1[N_]; // SoA bboxes (3.2 KB)
    __shared__ int   s_tl[N_];                                   // labels     (0.8 KB)
    __shared__ float s_lg[QT][C_];           // this block's 8 logit rows      (2.9 KB)

    const int b    = blockIdx.y;
    const int q0   = blockIdx.x * QT;
    const int tid  = threadIdx.x;
    const int wave = tid >> 5;
    const int lane = tid & 31;

    // ---- stage shared-target tile + logit rows with async tensor-path loads ----
    {   // polylines: 200 rows x 8 float4 chunks, padded LDS destination
        const float* gp = tgt_poly + (size_t)b * N_ * TWO_K;
        const uint32_t lbase = (uint32_t)(uintptr_t)s_tp;
        for (int i = tid; i < N_ * 8; i += BLOCK) {
            const int row = i >> 3, j4 = i & 7;
            async_ld_b128(lbase + (uint32_t)(row * TP_STRIDE * 4 + j4 * 16),
                          (uint32_t)(row * 128 + j4 * 16), gp);
        }
    }
    {   // labels
        const int* gl = tgt_labels + (size_t)b * N_;
        const uint32_t lbase = (uint32_t)(uintptr_t)s_tl;
        if (tid < N_) async_ld_b32(lbase + (uint32_t)(tid * 4), (uint32_t)(tid * 4), gl);
    }
    {   // logits for the 8 queries owned by this block
        const float* gg = pred_logits + ((size_t)b * Q_ + q0) * C_;
        const uint32_t lbase = (uint32_t)(uintptr_t)(&s_lg[0][0]);
        for (int i = tid; i < QT * C_; i += BLOCK) {
            const int row = i / C_;
            if (q0 + row < Q_)
                async_ld_b32(lbase + (uint32_t)(i * 4), (uint32_t)(i * 4), gg);
        }
    }
    wait_async_zero();
    __syncthreads();

    // ---- target bboxes from the staged tile ----
    for (int n = tid; n < N_; n += BLOCK) {
        const float* p = &s_tp[n * TP_STRIDE];
        float x0 = p[0], y0 = p[1], x1 = p[0], y1 = p[1];
        #pragma unroll
        for (int k = 1; k < K_; ++k) {
            const float x = p[2 * k], y = p[2 * k + 1];
            x0 = fminf(x0, x); x1 = fmaxf(x1, x);
            y0 = fminf(y0, y); y1 = fmaxf(y1, y);
        }
        s_bx0[n] = x0; s_by0[n] = y0; s_bx1[n] = x1; s_by1[n] = y1;
    }
    __syncthreads();

    const int q = q0 + wave;
    if (q >= Q_) return;   // no barriers past this point

    // ---- per-wave query state (wave-uniform) ----
    float a[TWO_K];
    {
        const float* pp = pred_poly + ((size_t)b * Q_ + q) * TWO_K;
        #pragma unroll
        for (int j = 0; j < TWO_K; ++j) a[j] = pp[j];
    }
    float px0 = a[0], py0 = a[1], px1 = a[0], py1 = a[1];
    #pragma unroll
    for (int k = 1; k < K_; ++k) {
        px0 = fminf(px0, a[2 * k]);     px1 = fmaxf(px1, a[2 * k]);
        py0 = fminf(py0, a[2 * k + 1]); py1 = fmaxf(py1, a[2 * k + 1]);
    }
    const float area1 = fmaxf(px1 - px0, 0.f) * fmaxf(py1 - py0, 0.f);

    // softmax stats over 92 classes, 32-lane butterfly reductions (wave32)
    float m = -INFINITY;
    for (int c = lane; c < C_; c += 32) m = fmaxf(m, s_lg[wave][c]);
    #pragma unroll
    for (int off = 16; off > 0; off >>= 1) m = fmaxf(m, __shfl_xor(m, off, 32));
    float ssum = 0.f;
    for (int c = lane; c < C_; c += 32) ssum += expf(s_lg[wave][c] - m);
    #pragma unroll
    for (int off = 16; off > 0; off >>= 1) ssum += __shfl_xor(ssum, off, 32);
    const float inv_sum = 1.f / ssum;

    // ---- lanes sweep the 200 targets ----
    for (int n = lane; n < N_; n += 32) {
        float t[TWO_K];
        #pragma unroll
        for (int j = 0; j < TWO_K; ++j) t[j] = s_tp[n * TP_STRIDE + j];

        float fwd = 0.f, rev = 0.f;
        #pragma unroll
        for (int p = 0; p < K_; ++p) {   // both orientations from the same registers
            fwd += fabsf(a[2 * p]     - t[2 * p])
                 + fabsf(a[2 * p + 1] - t[2 * p + 1]);
            rev += fabsf(a[2 * p]     - t[2 * (K_ - 1 - p)])
                 + fabsf(a[2 * p + 1] - t[2 * (K_ - 1 - p) + 1]);
        }
        const float cpoly = fminf(fwd, rev) * (1.f / (float)TWO_K);

        const float tx0 = s_bx0[n], ty0 = s_by0[n], tx1 = s_bx1[n], ty1 = s_by1[n];
        const float area2 = fmaxf(tx1 - tx0, 0.f) * fmaxf(ty1 - ty0, 0.f);
        const float iw = fminf(px1, tx1) - fmaxf(px0, tx0);
        const float ih = fminf(py1, ty1) - fmaxf(py0, ty0);
        const float inter = fmaxf(iw, 0.f) * fmaxf(ih, 0.f);
        const float uni = area1 + area2 - inter;
        const float iou = inter / fmaxf(uni, EPS_);
        const float cw = fmaxf(px1, tx1) - fminf(px0, tx0);
        const float ch = fmaxf(py1, ty1) - fminf(py0, ty0);
        const float areac = fmaxf(cw, 0.f) * fmaxf(ch, 0.f);
        const float giou = iou - (areac - uni) / fmaxf(areac, EPS_);

        const int lbl = s_tl[n];
        const float cclass = -expf(s_lg[wave][lbl] - m) * inv_sum;

        out[((size_t)b * Q_ + q) * N_ + n] = cclass + 5.f * cpoly - giou;
    }
}

extern "C" void kernel_launch(void* const* d_in, const int* in_sizes, int n_in,
                              void* d_out, int out_size, void* d_ws, size_t ws_size,
                              hipStream_t stream) {
    (void)in_sizes; (void)n_in; (void)out_size; (void)d_ws; (void)ws_size;
    const float* pred_logits = (const float*)d_in[0];
    const float* pred_poly   = (const float*)d_in[1];
    const int*   tgt_labels  = (const int*)d_in[2];
    const float* tgt_poly    = (const float*)d_in[3];
    float* out = (float*)d_out;

    dim3 grid((Q_ + QT - 1) / QT, B_);   // 63 x 32 blocks
    dim3 block(BLOCK);                   // 8 wave32s
    HungarianMatcherPolyline_kernel<<<grid, block, 0, stream>>>(
        pred_logits, pred_poly, tgt_labels, tgt_poly, out);
}